// TopLeftPool_19902878449855
// MI455X (gfx1250) — compile-verified
//
#include <hip/hip_runtime.h>
#include <stdint.h>

// TopLeftPool: out[n,c,i,j] = x[n,c, argmax_{i'<=i, j'<=j} guide[n,c,i',j']]
// (ties: rightmost column wins within a row, bottom-most row wins across rows,
//  matching the reference's two `vb >= va` associative scans).
//
// One wave32 per 128x128 plane. The x-plane is staged into LDS with CDNA5
// async global->LDS copies (ASYNCcnt / s_wait_asynccnt), pipelined 16 rows
// ahead of the data-dependent gather.

#define HH 128
#define WW 128
#define PLANE (HH * WW)   // 16384 elements, 64 KiB fp32
#define LEAD 16           // async staging lead distance (rows)

// Async copy of 512B (one row: 16B per lane * 32 lanes) global -> LDS.
// vdst = per-lane LDS byte address, vaddr = per-lane 64-bit global address.
__device__ __forceinline__ void async_row_to_lds(const float* gsrc, uint32_t lds_byte_off) {
    uint64_t ga = (uint64_t)(uintptr_t)gsrc;
    asm volatile("global_load_async_to_lds_b128 %0, %1, off"
                 :: "v"(lds_byte_off), "v"(ga)
                 : "memory");
}
// Per-wave async loads complete in order, so waiting ASYNCcnt<=LEAD after
// issuing row i+LEAD guarantees rows <= i have landed in LDS.
__device__ __forceinline__ void wait_async_le_lead() {
    asm volatile("s_wait_asynccnt 0x10" ::: "memory");
}
__device__ __forceinline__ void wait_async_zero() {
    asm volatile("s_wait_asynccnt 0x0" ::: "memory");
}

__global__ __launch_bounds__(32)
void topleft_pool_kernel(const float* __restrict__ x,
                         const float* __restrict__ guide,
                         float* __restrict__ out)
{
    __shared__ float sx[PLANE];                 // staged x plane (64 KiB)

    const int p    = blockIdx.x;                // plane = (n,c)
    const int lane = threadIdx.x & 31;
    const size_t pbase = (size_t)p * PLANE;
    const float* xp = x     + pbase;
    const float* gp = guide + pbase;
    float*       op = out   + pbase;

    // Flat shared pointer truncated to 32 bits == wave-relative LDS byte addr
    // (generic LDS addresses are {aperture, offset[31:0]}).
    const uint32_t lds0 = (uint32_t)(uintptr_t)(&sx[0]) + (uint32_t)lane * 16u;

    // Prologue: stage x rows [0, LEAD).
#pragma unroll
    for (int r = 0; r < LEAD; ++r)
        async_row_to_lds(xp + r * WW + lane * 4, lds0 + (uint32_t)r * (WW * 4));

    // Running column state (H-direction scan), 4 columns per lane.
    const float NEG = -__builtin_inff();
    float rv0 = NEG, rv1 = NEG, rv2 = NEG, rv3 = NEG;
    int   ri0 = 0,   ri1 = 0,   ri2 = 0,   ri3 = 0;

    float4 gcur = *(const float4*)(gp + lane * 4);      // guide row 0

    for (int i = 0; i < HH; ++i) {
        // Keep the async pipeline LEAD rows ahead; ensure rows <= i staged.
        if (i + LEAD < HH) {
            async_row_to_lds(xp + (i + LEAD) * WW + lane * 4,
                             lds0 + (uint32_t)(i + LEAD) * (WW * 4));
            wait_async_le_lead();
        } else {
            wait_async_zero();
        }

        // Software-pipelined guide load for the next row (independent of
        // this row's compute; compiler overlaps it with the scan).
        float4 gnext = gcur;
        if (i + 1 < HH) gnext = *(const float4*)(gp + (i + 1) * WW + lane * 4);

        // ---- W-direction: 4-wide serial inclusive scan (later wins ties) ----
        const int base = i * WW + lane * 4;
        float lv0 = gcur.x;                       int li0 = base;
        float lv1; int li1;
        if (gcur.y >= lv0) { lv1 = gcur.y; li1 = base + 1; } else { lv1 = lv0; li1 = li0; }
        float lv2; int li2;
        if (gcur.z >= lv1) { lv2 = gcur.z; li2 = base + 2; } else { lv2 = lv1; li2 = li1; }
        float lv3; int li3;
        if (gcur.w >= lv2) { lv3 = gcur.w; li3 = base + 3; } else { lv3 = lv2; li3 = li2; }

        // ---- wave32 inclusive scan of lane totals (5 shuffle steps) ----
        float sv = lv3; int si = li3;
#pragma unroll
        for (int d = 1; d < 32; d <<= 1) {
            float ov = __shfl_up(sv, d, 32);
            int   oi = __shfl_up(si, d, 32);
            if (lane >= d && !(sv >= ov)) { sv = ov; si = oi; }   // comb(earlier, later)
        }
        // Exclusive prefix over lanes < lane.
        float pv = __shfl_up(sv, 1, 32);
        int   pi = __shfl_up(si, 1, 32);
        if (lane == 0) { pv = NEG; pi = 0; }

        // Row-prefix (value,idx) for this lane's 4 columns: local wins ties.
        float q0, q1, q2, q3; int j0, j1, j2, j3;
        if (lv0 >= pv) { q0 = lv0; j0 = li0; } else { q0 = pv; j0 = pi; }
        if (lv1 >= pv) { q1 = lv1; j1 = li1; } else { q1 = pv; j1 = pi; }
        if (lv2 >= pv) { q2 = lv2; j2 = li2; } else { q2 = pv; j2 = pi; }
        if (lv3 >= pv) { q3 = lv3; j3 = li3; } else { q3 = pv; j3 = pi; }

        // ---- H-direction running combine: current (later) row wins ties ----
        if (q0 >= rv0) { rv0 = q0; ri0 = j0; }
        if (q1 >= rv1) { rv1 = q1; ri1 = j1; }
        if (q2 >= rv2) { rv2 = q2; ri2 = j2; }
        if (q3 >= rv3) { rv3 = q3; ri3 = j3; }

        // Gather x at the rectangle argmax from LDS; coalesced row store.
        float4 o;
        o.x = sx[ri0]; o.y = sx[ri1]; o.z = sx[ri2]; o.w = sx[ri3];
        *(float4*)(op + base) = o;

        gcur = gnext;
    }
}

extern "C" void kernel_launch(void* const* d_in, const int* in_sizes, int n_in,
                              void* d_out, int out_size, void* d_ws, size_t ws_size,
                              hipStream_t stream) {
    (void)in_sizes; (void)n_in; (void)d_ws; (void)ws_size;
    const float* x     = (const float*)d_in[0];
    const float* guide = (const float*)d_in[1];
    float*       out   = (float*)d_out;

    const int nplanes = out_size / PLANE;       // 8*256 = 2048 planes
    topleft_pool_kernel<<<dim3(nplanes), dim3(32), 0, stream>>>(x, guide, out);
}